// KGCN_78520592105834
// MI455X (gfx1250) — compile-verified
//
#include <hip/hip_runtime.h>
#include <hip/hip_bf16.h>
#include <math.h>

// KGCN forward for MI455X (gfx1250, wave32).
// Inputs (setup_inputs order):
//  0 pairs        (B,2) int32
//  1 adj_entity   (N_ENTITY,16) int32
//  2 adj_relation (N_ENTITY,16) int32
//  3 entity_emb   (N_ENTITY,64) f32
//  4 relation_emb (32,64) f32
//  5 user_emb     (N_USER,64) f32
//  6 W            (64,64) f32
//  7 b            (64,) f32
// Output: (B,) f32 sigmoid scores.
// Workspace: ev1_new = B*16*64 f32 (hop-1 updated neighbor embeddings).

typedef __attribute__((ext_vector_type(2))) float v2f;
typedef __attribute__((ext_vector_type(8))) float v8f;

#define DIM 64
#define KN 16
#define LSTR 66   // padded LDS row stride (floats): 2*row bank offset -> conflict-free column reads

// ---- helpers -------------------------------------------------------------

__device__ __forceinline__ float wave_allsum(float p) {
    // full 32-lane butterfly: result replicated in every lane
    #pragma unroll
    for (int off = 16; off >= 1; off >>= 1) p += __shfl_xor(p, off, 32);
    return p;
}

// One 16x16 output tile of H = X(16x64) * W^T(64x64), X in LDS (stride LSTR),
// Wlds holds W row-major with stride LSTR (so B-matrix element (K,N) = Wlds[N*LSTR+K]).
__device__ __forceinline__ v8f wmma_tile_f32(const float* Xs, const float* Wlds,
                                             int n0, int lane) {
    const int half = lane >> 4;      // K sub-slot: lanes 0-15 hold K {0,1}, 16-31 hold {2,3}
    const int lr   = lane & 15;      // A row / B column within tile
    v8f acc = {};
    #pragma unroll
    for (int kc = 0; kc < 16; ++kc) {
        const int kb = kc * 4 + 2 * half;
        v2f a, bmat;
        a.x    = Xs[lr * LSTR + kb];
        a.y    = Xs[lr * LSTR + kb + 1];
        bmat.x = Wlds[(n0 + lr) * LSTR + kb];
        bmat.y = Wlds[(n0 + lr) * LSTR + kb + 1];
        acc = __builtin_amdgcn_wmma_f32_16x16x4_f32(
            /*neg_a=*/false, a, /*neg_b=*/false, bmat,
            /*c_mod=*/(short)0, acc, /*reuse_a=*/false, /*reuse_b=*/false);
    }
    return acc;
}

// ---- kernel 1: hop-1 neighbor update (layer0, i=1) -----------------------
// grid = B blocks, block = 512 threads (16 waves); wave m handles neighbor m.

__global__ __launch_bounds__(512)
void kgcn_hop1(const int* __restrict__ pairs,
               const int* __restrict__ adj_entity,
               const int* __restrict__ adj_relation,
               const float* __restrict__ entity_emb,
               const float* __restrict__ relation_emb,
               const float* __restrict__ user_emb,
               const float* __restrict__ W,
               const float* __restrict__ bias,
               float* __restrict__ ev1) {
    __shared__ float Xs[16 * LSTR];
    __shared__ float Ws[64 * LSTR];

    const int b    = blockIdx.x;
    const int tid  = threadIdx.x;
    const int wave = tid >> 5;
    const int lane = tid & 31;

    // stage W into LDS (padded stride)
    for (int i = tid; i < 64 * 64; i += 512)
        Ws[(i >> 6) * LSTR + (i & 63)] = W[i];

    const int user = pairs[2 * b];
    const int item = pairs[2 * b + 1];
    const float u0 = user_emb[user * DIM + 2 * lane];
    const float u1 = user_emb[user * DIM + 2 * lane + 1];

    const int m  = wave;
    const int e1 = adj_entity[item * KN + m];

    int nb_e = 0, nb_r = 0;
    if (lane < KN) {
        nb_e = adj_entity[e1 * KN + lane];
        nb_r = adj_relation[e1 * KN + lane];
    }

    // attention scores: u . relation_emb[rel_k]  (coalesced 2 floats/lane)
    float s[KN];
    #pragma unroll
    for (int k = 0; k < KN; ++k) {
        const int rid = __shfl(nb_r, k, 32);
        const float r0 = relation_emb[rid * DIM + 2 * lane];
        const float r1 = relation_emb[rid * DIM + 2 * lane + 1];
        s[k] = wave_allsum(u0 * r0 + u1 * r1);
    }
    // softmax over K (replicated per lane)
    float mx = s[0];
    #pragma unroll
    for (int k = 1; k < KN; ++k) mx = fmaxf(mx, s[k]);
    float ssum = 0.f;
    #pragma unroll
    for (int k = 0; k < KN; ++k) { s[k] = __expf(s[k] - mx); ssum += s[k]; }
    const float inv = 1.f / ssum;

    // weighted sum of hop-2 entity rows (L2-resident gather, coalesced)
    float a0 = 0.f, a1 = 0.f;
    #pragma unroll
    for (int k = 0; k < KN; ++k) {
        const int eid = __shfl(nb_e, k, 32);
        if (k + 1 < KN) {
            const int enx = __shfl(nb_e, k + 1, 32);
            __builtin_prefetch(&entity_emb[enx * DIM + 2 * lane], 0, 1);
        }
        const float w = s[k] * inv;
        a0 = fmaf(w, entity_emb[eid * DIM + 2 * lane], a0);
        a1 = fmaf(w, entity_emb[eid * DIM + 2 * lane + 1], a1);
    }
    // x = self + aggregated neighbors
    Xs[m * LSTR + 2 * lane]     = entity_emb[e1 * DIM + 2 * lane] + a0;
    Xs[m * LSTR + 2 * lane + 1] = entity_emb[e1 * DIM + 2 * lane + 1] + a1;
    __syncthreads();

    // H = relu(X @ W^T + bias); waves 0..3 each compute one 16x16 output tile
    if (wave < 4) {
        const int n0  = wave * 16;
        const int half = lane >> 4;
        const int lr   = lane & 15;
        v8f acc = wmma_tile_f32(Xs, Ws, n0, lane);
        #pragma unroll
        for (int j = 0; j < 8; ++j) {
            const int row = j + 8 * half;
            const int col = n0 + lr;
            const float h = fmaxf(acc[j] + bias[col], 0.f);
            ev1[((size_t)b * 16 + row) * DIM + col] = h;
        }
    }
}

// ---- kernel 2: item-level layers (layer0 i=0, layer1 i=0) + score --------
// grid = B/16 blocks, block = 512 threads; wave w = batch element b0+w.

__global__ __launch_bounds__(512)
void kgcn_final(const int* __restrict__ pairs,
                const int* __restrict__ adj_entity,
                const int* __restrict__ adj_relation,
                const float* __restrict__ entity_emb,
                const float* __restrict__ relation_emb,
                const float* __restrict__ user_emb,
                const float* __restrict__ W,
                const float* __restrict__ bias,
                const float* __restrict__ ev1,
                float* __restrict__ out,
                int Btot) {
    __shared__ float Xs[16 * LSTR];
    __shared__ float Hs[16 * LSTR];
    __shared__ float Ws[64 * LSTR];

    const int tid  = threadIdx.x;
    const int wave = tid >> 5;
    const int lane = tid & 31;
    int b = blockIdx.x * 16 + wave;
    if (b >= Btot) b = Btot - 1;   // uniform per-wave clamp, keeps barriers convergent

    for (int i = tid; i < 64 * 64; i += 512)
        Ws[(i >> 6) * LSTR + (i & 63)] = W[i];

    const int user = pairs[2 * b];
    const int item = pairs[2 * b + 1];
    const float u0 = user_emb[user * DIM + 2 * lane];
    const float u1 = user_emb[user * DIM + 2 * lane + 1];

    int nb_e = 0, nb_r = 0;
    if (lane < KN) {
        nb_e = adj_entity[item * KN + lane];
        nb_r = adj_relation[item * KN + lane];
    }

    // item-level attention (same weights used in both layers)
    float s[KN];
    #pragma unroll
    for (int k = 0; k < KN; ++k) {
        const int rid = __shfl(nb_r, k, 32);
        const float r0 = relation_emb[rid * DIM + 2 * lane];
        const float r1 = relation_emb[rid * DIM + 2 * lane + 1];
        s[k] = wave_allsum(u0 * r0 + u1 * r1);
    }
    float mx = s[0];
    #pragma unroll
    for (int k = 1; k < KN; ++k) mx = fmaxf(mx, s[k]);
    float ssum = 0.f;
    #pragma unroll
    for (int k = 0; k < KN; ++k) { s[k] = __expf(s[k] - mx); ssum += s[k]; }
    const float inv = 1.f / ssum;

    // layer0 i=0 aggregation: original first-hop embeddings
    float a0 = 0.f, a1 = 0.f;
    // layer1 i=0 aggregation: hop-1 updated embeddings from kernel 1
    float n20 = 0.f, n21 = 0.f;
    #pragma unroll
    for (int k = 0; k < KN; ++k) {
        const int eid = __shfl(nb_e, k, 32);
        const float w = s[k] * inv;
        a0  = fmaf(w, entity_emb[eid * DIM + 2 * lane], a0);
        a1  = fmaf(w, entity_emb[eid * DIM + 2 * lane + 1], a1);
        n20 = fmaf(w, ev1[((size_t)b * 16 + k) * DIM + 2 * lane], n20);
        n21 = fmaf(w, ev1[((size_t)b * 16 + k) * DIM + 2 * lane + 1], n21);
    }
    Xs[wave * LSTR + 2 * lane]     = entity_emb[item * DIM + 2 * lane] + a0;
    Xs[wave * LSTR + 2 * lane + 1] = entity_emb[item * DIM + 2 * lane + 1] + a1;
    __syncthreads();

    // matmul 1: H = relu(X @ W^T + bias)
    if (wave < 4) {
        const int n0  = wave * 16;
        const int half = lane >> 4;
        const int lr   = lane & 15;
        v8f acc = wmma_tile_f32(Xs, Ws, n0, lane);
        #pragma unroll
        for (int j = 0; j < 8; ++j) {
            const int row = j + 8 * half;
            const int col = n0 + lr;
            Hs[row * LSTR + col] = fmaxf(acc[j] + bias[col], 0.f);
        }
    }
    __syncthreads();

    // x2 = H + aggregated(updated neighbors)
    Xs[wave * LSTR + 2 * lane]     = Hs[wave * LSTR + 2 * lane] + n20;
    Xs[wave * LSTR + 2 * lane + 1] = Hs[wave * LSTR + 2 * lane + 1] + n21;
    __syncthreads();

    // matmul 2: H2 = X2 @ W^T + bias (tanh applied after)
    if (wave < 4) {
        const int n0  = wave * 16;
        const int half = lane >> 4;
        const int lr   = lane & 15;
        v8f acc = wmma_tile_f32(Xs, Ws, n0, lane);
        #pragma unroll
        for (int j = 0; j < 8; ++j) {
            const int row = j + 8 * half;
            const int col = n0 + lr;
            Hs[row * LSTR + col] = acc[j] + bias[col];
        }
    }
    __syncthreads();

    // score = sigmoid(u . tanh(h2))
    const float t0 = tanhf(Hs[wave * LSTR + 2 * lane]);
    const float t1 = tanhf(Hs[wave * LSTR + 2 * lane + 1]);
    const float dot = wave_allsum(u0 * t0 + u1 * t1);
    if (lane == 0) out[b] = 1.f / (1.f + __expf(-dot));
}

// ---- launch --------------------------------------------------------------

extern "C" void kernel_launch(void* const* d_in, const int* in_sizes, int n_in,
                              void* d_out, int out_size, void* d_ws, size_t ws_size,
                              hipStream_t stream) {
    const int*   pairs        = (const int*)d_in[0];
    const int*   adj_entity   = (const int*)d_in[1];
    const int*   adj_relation = (const int*)d_in[2];
    const float* entity_emb   = (const float*)d_in[3];
    const float* relation_emb = (const float*)d_in[4];
    const float* user_emb     = (const float*)d_in[5];
    const float* W            = (const float*)d_in[6];
    const float* bias         = (const float*)d_in[7];
    float*       out          = (float*)d_out;
    float*       ev1          = (float*)d_ws;   // B*16*64 f32

    const int B = in_sizes[0] / 2;

    kgcn_hop1<<<B, 512, 0, stream>>>(pairs, adj_entity, adj_relation, entity_emb,
                                     relation_emb, user_emb, W, bias, ev1);
    kgcn_final<<<(B + 15) / 16, 512, 0, stream>>>(pairs, adj_entity, adj_relation,
                                                  entity_emb, relation_emb, user_emb,
                                                  W, bias, ev1, out, B);
}